// S4DReal_71425306133128
// MI455X (gfx1250) — compile-verified
//
#include <hip/hip_runtime.h>

// S4D-real: y = causal_conv(x, K) with K[l,c] = sum_s (C*Bd)[c,s] * r[c,s]^l.
// Step 1 builds K via per-state geometric recurrences (no LxCxS tensor).
// Step 2 does the per-channel causal convolution as a block-Toeplitz matmul
// using V_WMMA_F32_16X16X4_F32 chains (fp32-exact path on CDNA5).
// Round-2 refinements: K staged *reversed* in LDS so A-fragment pairs are
// ascending (kills the v_dual_mov operand swaps before each WMMA), and
// group->wave assignment {w, 7-w, 8+w, 15-w} for exact load balance.

#define L_LEN 4096
#define CHANS 512
#define S_DIM 64
#define DT_SR (1.0f / 4096.0f)

typedef __attribute__((ext_vector_type(2))) float v2f;
typedef __attribute__((ext_vector_type(8))) float v8f;

// ---------------------------------------------------------------------------
// Kernel 1: build Kt[c][l] (transposed kernel, contiguous per channel).
// Thread = (channel, 32-long l-chunk). 512*128 = 65536 threads.
// ---------------------------------------------------------------------------
__global__ void __launch_bounds__(256) s4d_build_kernel(
    const float* __restrict__ lognegA,  // [CHANS, S_DIM]
    const float* __restrict__ Bmat,     // [CHANS, S_DIM]
    const float* __restrict__ Cmat,     // [CHANS, S_DIM]
    float* __restrict__ Kt)             // [CHANS, L_LEN]
{
    const int tid   = blockIdx.x * 256 + threadIdx.x;
    const int c     = tid >> 7;          // channel 0..511
    const int chunk = tid & 127;         // l-chunk 0..127
    const int l0    = chunk * 32;

    float acc[32];
#pragma unroll
    for (int i = 0; i < 32; ++i) acc[i] = 0.0f;

    for (int sb = 0; sb < S_DIM; sb += 8) {
        float r[8], w[8], p[8];
#pragma unroll
        for (int u = 0; u < 8; ++u) {
            const int idx = c * S_DIM + sb + u;
            const float A  = -__expf(lognegA[idx]);      // strictly negative
            const float rr = __expf(A * DT_SR);          // discrete decay
            const float Bd = (rr - 1.0f) * Bmat[idx] / A;
            r[u] = rr;
            w[u] = Cmat[idx] * Bd;
            p[u] = __expf((float)l0 * DT_SR * A);        // r^l0 seed
        }
#pragma unroll
        for (int i = 0; i < 32; ++i) {
#pragma unroll
            for (int u = 0; u < 8; ++u) {
                acc[i] = fmaf(w[u], p[u], acc[i]);
                p[u] *= r[u];
            }
        }
    }

    float* dst = Kt + (size_t)c * L_LEN + l0;
#pragma unroll
    for (int i = 0; i < 32; ++i) dst[i] = acc[i];
}

// ---------------------------------------------------------------------------
// Kernel 2: per-channel causal conv as block-Toeplitz matmul via WMMA f32.
//   Y[t, b] = sum_k T_k[t, j] * X[j, b-k],  T_k[t,j] = K[16k + t - j]
// One workgroup (4 waves / 128 threads) per channel.
//  - xs: x[:,c] with 256 zero floats in front (covers block index down to -15)
//  - ksr: K[:,c] stored REVERSED, ksr[4095 - l] = K[l], with a 16-float zero
//    tail covering l in [-15,-1]. A-fragment reads K[q] and K[q-1], which in
//    reversed storage are ascending addresses -> single merged LDS pair load,
//    no operand-swap movs. Branch-free boundaries keep EXEC all-ones at WMMA.
// ---------------------------------------------------------------------------
#define XPAD 256

__global__ void __launch_bounds__(128) s4d_toeplitz_conv(
    const float* __restrict__ x,    // [L_LEN, CHANS]
    const float* __restrict__ Kt,   // [CHANS, L_LEN]
    float* __restrict__ y)          // [L_LEN, CHANS]
{
    __shared__ float xs[XPAD + L_LEN];
    __shared__ float ksr[L_LEN + 16];   // reversed K + zero tail (l = -1..-15)

    const int c   = blockIdx.x;
    const int tix = threadIdx.x;    // 0..127

    for (int i = tix; i < XPAD; i += 128) xs[i] = 0.0f;
    if (tix < 16) ksr[L_LEN + tix] = 0.0f;
    for (int l = tix; l < L_LEN; l += 128) {
        xs[XPAD + l]          = x[(size_t)l * CHANS + c];   // strided; L2-hot
        ksr[(L_LEN - 1) - l]  = Kt[(size_t)c * L_LEN + l];  // contiguous read
    }
    __syncthreads();

    const int wave = tix >> 5;
    const int lane = tix & 31;
    const int n    = lane & 15;     // A row (t), B/D column (i')
    const int h    = lane >> 4;     // selects K-dim pair {2h, 2h+1}

    // Balanced group sets: wave w handles {w, 7-w, 8+w, 15-w} -> each wave
    // does exactly 544 lag iterations.
    const int gtab[4] = { wave, 7 - wave, 8 + wave, 15 - wave };

    for (int gg = 0; gg < 4; ++gg) {
        const int G = gtab[gg];                 // group of 16 output blocks
        v8f acc = {0.0f, 0.0f, 0.0f, 0.0f, 0.0f, 0.0f, 0.0f, 0.0f};

        const int kmax = 16 * G + 15;           // causal lag bound for group
        for (int k = 0; k <= kmax; ++k) {
            // A[t][j] = K[16k + t - j]; reversed addr = 4095 - (16k + t) + j
            const int arbase = (L_LEN - 1) - 16 * k - n;
            const int bbase  = XPAD + 16 * (16 * G + n - k);  // x block b-k
#pragma unroll
            for (int kk = 0; kk < 4; ++kk) {     // 16-wide j contraction, K=4
                const int j0 = 4 * kk + 2 * h;
                v2f a, b;
                a.x = ksr[arbase + j0];          // K[16k + t - j0]
                a.y = ksr[arbase + j0 + 1];      // K[16k + t - (j0+1)]
                b.x = xs[bbase + j0];            // B[j0][i']
                b.y = xs[bbase + j0 + 1];        // B[j0+1][i']
                acc = __builtin_amdgcn_wmma_f32_16x16x4_f32(
                    /*neg_a=*/false, a, /*neg_b=*/false, b,
                    /*c_mod=*/(short)0, acc,
                    /*reuse_a=*/false, /*reuse_b=*/false);
            }
        }

        // D layout: VGPR v -> row (v + 8h), lane&15 -> column i'.
        const int ob = 16 * G + n;               // output block index
#pragma unroll
        for (int v = 0; v < 8; ++v) {
            const int row = 16 * ob + v + 8 * h;
            y[(size_t)row * CHANS + c] = acc[v];
        }
    }
}

// ---------------------------------------------------------------------------
// Harness entry. d_in order: x, lognegA, B, C. d_ws holds Kt (8 MB).
// ---------------------------------------------------------------------------
extern "C" void kernel_launch(void* const* d_in, const int* in_sizes, int n_in,
                              void* d_out, int out_size, void* d_ws, size_t ws_size,
                              hipStream_t stream) {
    (void)in_sizes; (void)n_in; (void)out_size; (void)ws_size;
    const float* x       = (const float*)d_in[0];
    const float* lognegA = (const float*)d_in[1];
    const float* Bmat    = (const float*)d_in[2];
    const float* Cmat    = (const float*)d_in[3];
    float* y  = (float*)d_out;
    float* Kt = (float*)d_ws;   // CHANS * L_LEN * 4 = 8 MB scratch

    s4d_build_kernel<<<256, 256, 0, stream>>>(lognegA, Bmat, Cmat, Kt);
    s4d_toeplitz_conv<<<CHANS, 128, 0, stream>>>(x, Kt, y);
}